// TreeGRU_5798205849962
// MI455X (gfx1250) — compile-verified
//
#include <hip/hip_runtime.h>
#include <hip/hip_bf16.h>

typedef float v2f __attribute__((ext_vector_type(2)));
typedef float v8f __attribute__((ext_vector_type(8)));

#define N_NODES 65536
#define D 128

// ---------------------------------------------------------------------------
// 4 vertically-stacked 16x16 output tiles (rows m0..m0+63, cols n0..n0+15) of
// A[Nrows,128] @ B[128,128] via V_WMMA_F32_16X16X4_F32.
// The B fragment (2 K-rows x 16 cols) is loaded ONCE per k-step and reused by
// all 4 WMMAs -> 1.5 load-issues per WMMA instead of 3 (load-issue bound loop).
// Lane layout (ISA 7.12.2): lanes 0-15 hold K=kb+0,kb+1; lanes 16-31 K=kb+2,kb+3;
// C/D: c[v] = D[m0 + v + half*8][n0 + (lane&15)].
// ---------------------------------------------------------------------------
__device__ __forceinline__ void gemm_tile4(const float* __restrict__ A,
                                           const float* __restrict__ B,
                                           int m0, int n0,
                                           v8f& c0, v8f& c1, v8f& c2, v8f& c3) {
  const int lane = threadIdx.x & 31;
  const int half = lane >> 4;
  const int r    = lane & 15;
  const int koff = half * 2;
  const float* __restrict__ a0 = A + (size_t)(m0 + r) * D + koff;
  const float* __restrict__ a1 = a0 + 16 * D;
  const float* __restrict__ a2 = a0 + 32 * D;
  const float* __restrict__ a3 = a0 + 48 * D;
  const float* __restrict__ bcol = B + n0 + r;
#pragma unroll
  for (int kb = 0; kb < D; kb += 4) {
    v2f b;
    b.x = bcol[(size_t)(kb + koff) * D];
    b.y = bcol[(size_t)(kb + koff + 1) * D];
    v2f va0 = *reinterpret_cast<const v2f*>(a0 + kb);   // 8B aligned (kb+koff even)
    v2f va1 = *reinterpret_cast<const v2f*>(a1 + kb);
    v2f va2 = *reinterpret_cast<const v2f*>(a2 + kb);
    v2f va3 = *reinterpret_cast<const v2f*>(a3 + kb);
    c0 = __builtin_amdgcn_wmma_f32_16x16x4_f32(false, va0, false, b, (short)0, c0, false, false);
    c1 = __builtin_amdgcn_wmma_f32_16x16x4_f32(false, va1, false, b, (short)0, c1, false, false);
    c2 = __builtin_amdgcn_wmma_f32_16x16x4_f32(false, va2, false, b, (short)0, c2, false, false);
    c3 = __builtin_amdgcn_wmma_f32_16x16x4_f32(false, va3, false, b, (short)0, c3, false, false);
  }
}

// ---------------------------------------------------------------------------
__global__ void __launch_bounds__(256) zero_kernel(float4* __restrict__ p, int n4) {
  int i = blockIdx.x * blockDim.x + threadIdx.x;
  if (i < n4) p[i] = make_float4(0.f, 0.f, 0.f, 0.f);
}

// rsig = sigmoid(f_dst @ wr + h @ ur + br)   (node-level, independent of edge pass)
// One wave per 64-row x 16-col output slab.
__global__ void __launch_bounds__(256)
rsig_kernel(const float* __restrict__ f_dst, const float* __restrict__ h,
            const float* __restrict__ wr, const float* __restrict__ ur,
            const float* __restrict__ br, float* __restrict__ rsig) {
  const int wave = blockIdx.x * (blockDim.x >> 5) + (threadIdx.x >> 5);
  const int m0 = (wave >> 3) << 6;   // 64-row slab
  const int n0 = (wave & 7) << 4;    // 8 column tiles
  v8f c[4] = {};
  gemm_tile4(f_dst, wr, m0, n0, c[0], c[1], c[2], c[3]);
  gemm_tile4(h,     ur, m0, n0, c[0], c[1], c[2], c[3]);
  const int lane = threadIdx.x & 31;
  const int half = lane >> 4;
  const int r    = lane & 15;
  const float bias = br[n0 + r];
#pragma unroll
  for (int t = 0; t < 4; ++t) {
#pragma unroll
    for (int v = 0; v < 8; ++v) {
      const int row = m0 + t * 16 + v + half * 8;
      const float x = c[t][v] + bias;
      rsig[(size_t)row * D + n0 + r] = 1.0f / (1.0f + __expf(-x));
    }
  }
}

// Fused edge pass: one wave per edge; s[dst] += h[src]; srth[dst] += rsig[src]*h[src].
// h / rsig / s / srth (128 MB total) are L2-resident -> atomics resolve in L2.
__global__ void __launch_bounds__(256)
edge_kernel(const float* __restrict__ h, const float* __restrict__ rsig,
            const int* __restrict__ src, const int* __restrict__ dst,
            float* __restrict__ s, float* __restrict__ srth, int E) {
  const int gid = blockIdx.x * blockDim.x + threadIdx.x;
  const int e = gid >> 5;
  if (e >= E) return;
  const int lane = threadIdx.x & 31;
  const int us = src[e];
  const int vd = dst[e];
  const float4 hv = reinterpret_cast<const float4*>(h    + (size_t)us * D)[lane];
  const float4 rv = reinterpret_cast<const float4*>(rsig + (size_t)us * D)[lane];
  float* sp = s    + (size_t)vd * D + lane * 4;
  float* rp = srth + (size_t)vd * D + lane * 4;
  atomicAdd(sp + 0, hv.x);        atomicAdd(sp + 1, hv.y);
  atomicAdd(sp + 2, hv.z);        atomicAdd(sp + 3, hv.w);
  atomicAdd(rp + 0, rv.x * hv.x); atomicAdd(rp + 1, rv.y * hv.y);
  atomicAdd(rp + 2, rv.z * hv.z); atomicAdd(rp + 3, rv.w * hv.w);
}

// z = sigmoid(f_src@wz + s@uz + bz); h~ = tanh(f_src@w + srth@u + b);
// out = (1-z)*s + z*h~     (one wave per 64x16 output slab, 8 accumulators)
__global__ void __launch_bounds__(256)
out_kernel(const float* __restrict__ f_src, const float* __restrict__ s,
           const float* __restrict__ srth,
           const float* __restrict__ wz, const float* __restrict__ uz,
           const float* __restrict__ bz,
           const float* __restrict__ w,  const float* __restrict__ u,
           const float* __restrict__ b,  float* __restrict__ out) {
  const int wave = blockIdx.x * (blockDim.x >> 5) + (threadIdx.x >> 5);
  const int m0 = (wave >> 3) << 6;
  const int n0 = (wave & 7) << 4;
  v8f cz[4] = {};
  v8f ch[4] = {};
  gemm_tile4(f_src, wz, m0, n0, cz[0], cz[1], cz[2], cz[3]);
  gemm_tile4(s,     uz, m0, n0, cz[0], cz[1], cz[2], cz[3]);
  gemm_tile4(f_src, w,  m0, n0, ch[0], ch[1], ch[2], ch[3]);
  gemm_tile4(srth,  u,  m0, n0, ch[0], ch[1], ch[2], ch[3]);
  const int lane = threadIdx.x & 31;
  const int half = lane >> 4;
  const int r    = lane & 15;
  const float bzv = bz[n0 + r];
  const float bv  = b[n0 + r];
#pragma unroll
  for (int t = 0; t < 4; ++t) {
#pragma unroll
    for (int v = 0; v < 8; ++v) {
      const int row = m0 + t * 16 + v + half * 8;
      const size_t idx = (size_t)row * D + n0 + r;
      const float z  = 1.0f / (1.0f + __expf(-(cz[t][v] + bzv)));
      const float ht = tanhf(ch[t][v] + bv);
      const float sv = s[idx];
      out[idx] = (1.0f - z) * sv + z * ht;
    }
  }
}

// ---------------------------------------------------------------------------
extern "C" void kernel_launch(void* const* d_in, const int* in_sizes, int n_in,
                              void* d_out, int out_size, void* d_ws, size_t ws_size,
                              hipStream_t stream) {
  const float* h     = (const float*)d_in[0];
  const float* f_src = (const float*)d_in[1];
  const float* f_dst = (const float*)d_in[2];
  const float* wz    = (const float*)d_in[3];
  const float* uz    = (const float*)d_in[4];
  const float* bz    = (const float*)d_in[5];
  const float* wr    = (const float*)d_in[6];
  const float* ur    = (const float*)d_in[7];
  const float* br    = (const float*)d_in[8];
  const float* w     = (const float*)d_in[9];
  const float* u     = (const float*)d_in[10];
  const float* b     = (const float*)d_in[11];
  const int*   src   = (const int*)d_in[12];
  const int*   dst   = (const int*)d_in[13];
  const int E = in_sizes[12];

  float* s    = (float*)d_ws;                       // [N,128]  32 MB
  float* srth = s    + (size_t)N_NODES * D;         // [N,128]  32 MB
  float* rsig = srth + (size_t)N_NODES * D;         // [N,128]  32 MB
  float* out  = (float*)d_out;

  // 1) zero the two segment-sum accumulators (ws is poisoned by harness)
  {
    const int n4 = 2 * N_NODES * D / 4;
    zero_kernel<<<(n4 + 255) / 256, 256, 0, stream>>>((float4*)s, n4);
  }
  // 2) node-level GEMM: rsig  (N/64 slabs x 8 col-tiles = 8192 waves)
  {
    const int waves = (N_NODES / 64) * (D / 16);
    rsig_kernel<<<waves / 8, 256, 0, stream>>>(f_dst, h, wr, ur, br, rsig);
  }
  // 3) fused edge scatter (one wave per edge)
  {
    const long long threads = (long long)E * 32;
    edge_kernel<<<(int)((threads + 255) / 256), 256, 0, stream>>>(h, rsig, src, dst,
                                                                  s, srth, E);
  }
  // 4) GEMMs + GRU epilogue
  {
    const int waves = (N_NODES / 64) * (D / 16);
    out_kernel<<<waves / 8, 256, 0, stream>>>(f_src, s, srth, wz, uz, bz, w, u, b, out);
  }
}